// AdversarialGeneratorv3_42949672960278
// MI455X (gfx1250) — compile-verified
//
#include <hip/hip_runtime.h>
#include <hip/hip_bf16.h>

typedef __attribute__((ext_vector_type(2))) float v2f;
typedef __attribute__((ext_vector_type(8))) float v8f;

#define B 8
#define NPT 4096
#define FIN 32
#define FDIM 33
#define FPAD 40          // padded feature dim: 10 WMMA steps of K=4
#define NSTEPS 10
#define KNN 32
#define ROWS 16          // rows per workgroup
#define LDS_STRIDE 4100  // 4096 + 4 to break bank-conflict pattern

// ---------------- Kernel 1: pad features, compute squared norms ----------------
__global__ void ama_prep_kernel(const float* __restrict__ x,
                                const float* __restrict__ noise,
                                float* __restrict__ fpad,
                                float* __restrict__ sq) {
  int p = blockIdx.x * blockDim.x + threadIdx.x;   // point id in [0, B*NPT)
  if (p >= B * NPT) return;
  const float* xi = x + (size_t)p * FIN;
  float* o = fpad + (size_t)p * FPAD;
  float s = 0.f;
#pragma unroll
  for (int d = 0; d < FIN; ++d) {
    float v = xi[d];
    o[d] = v;
    s += v * v;
  }
  float nz = noise[p];
  o[FIN] = nz;
  s += nz * nz;
#pragma unroll
  for (int d = FDIM; d < FPAD; ++d) o[d] = 0.f;
  sq[p] = s;
}

// ---------------- Kernel 2: WMMA distances + top-K + fused generator ----------------
__launch_bounds__(256, 1)
__global__ void ama_knn_main_kernel(const float* __restrict__ fpad,
                                    const float* __restrict__ sq,
                                    const float* __restrict__ y,
                                    const float* __restrict__ W1,   // [66][32] row-major
                                    const float* __restrict__ b1,   // [32]
                                    float* __restrict__ gen_out,    // [B][NPT][32]
                                    float* __restrict__ partials) { // [gridDim.x]
  __shared__ float d2s[ROWS * LDS_STRIDE];   // 16 x 4100 floats = 256.25 KB (CDNA5: 320KB/WG)
  __shared__ float errs[ROWS];

  const int b  = blockIdx.x >> 8;     // 256 row-blocks per batch
  const int rb = blockIdx.x & 255;
  const int i0 = rb * ROWS;
  const int tid  = threadIdx.x;
  const int wave = tid >> 5;
  const int lane = tid & 31;

  const float* fb  = fpad + (size_t)b * NPT * FPAD;
  const float* sqb = sq + (size_t)b * NPT;

  // ---- Stage 1: 16x4096 distance strip via V_WMMA_F32_16X16X4_F32 ----
  // fp32 16x4 A-fragment layout: lanes 0-15 hold M=lane K={0,1}; lanes 16-31 hold M=lane-16 K={2,3}.
  const int lr   = lane & 15;
  const int koff = (lane >> 4) << 1;
  const int mbase = (lane >> 4) << 3;

  v2f afrag[NSTEPS];
#pragma unroll
  for (int s = 0; s < NSTEPS; ++s)
    afrag[s] = *(const v2f*)(fb + (size_t)(i0 + lr) * FPAD + s * 4 + koff);

  float sqi[8];
#pragma unroll
  for (int r = 0; r < 8; ++r) sqi[r] = sqb[i0 + r + mbase];

  for (int jt = wave; jt < NPT / 16; jt += 8) {
    const int j0 = jt * 16;
    v8f c = {};
#pragma unroll
    for (int s = 0; s < NSTEPS; ++s) {
      // Gram = A * A^T: B-fragment (4x16, K-major) mirrors the A layout for the j-tile.
      v2f bf = *(const v2f*)(fb + (size_t)(j0 + lr) * FPAD + s * 4 + koff);
      c = __builtin_amdgcn_wmma_f32_16x16x4_f32(
          /*neg_a=*/false, afrag[s], /*neg_b=*/false, bf,
          /*c_mod=*/(short)0, c, /*reuse_a=*/false, /*reuse_b=*/false);
    }
    const int n = lane & 15;
    const int j = j0 + n;
    const float sqj = sqb[j];
#pragma unroll
    for (int r = 0; r < 8; ++r) {
      const int m = r + mbase;   // C/D layout: lanes<16 -> M=r, lanes>=16 -> M=r+8, N=lane&15
      d2s[m * LDS_STRIDE + j] = sqi[r] + sqj - 2.0f * c[r];
    }
  }
  __syncthreads();

  // ---- Stage 2+3: per-wave top-32 + softmax + weighted diff + generator ----
  const float INFV = 3.0e38f;
#pragma unroll 1
  for (int rr = 0; rr < 2; ++rr) {
    const int m = wave * 2 + rr;
    const int i = i0 + m;
    float* row = &d2s[m * LDS_STRIDE];

    // per-lane cached min over strided 128-element segment (j = lane + 32*t)
    float lmin = INFV; int lidx = 0;
    for (int t = 0; t < NPT / 32; ++t) {
      int j = lane + (t << 5);
      float v = row[j];
      if (v < lmin) { lmin = v; lidx = j; }
    }

    float myv = 0.f; int myidx = 0;   // lane k ends up holding k-th neighbor
#pragma unroll 1
    for (int k = 0; k < KNN; ++k) {
      float gv = lmin; int gi = lidx;
#pragma unroll
      for (int off = 16; off >= 1; off >>= 1) {
        float ov = __shfl_xor(gv, off, 32);
        int   oi = __shfl_xor(gi, off, 32);
        if (ov < gv || (ov == gv && oi < gi)) { gv = ov; gi = oi; }
      }
      if (lane == k) { myv = gv; myidx = gi; }
      if (lane == (gi & 31)) {             // owner lane extracts + rescans its segment
        row[gi] = INFV;
        lmin = INFV; lidx = 0;
        for (int t = 0; t < NPT / 32; ++t) {
          int j = lane + (t << 5);
          float v = row[j];
          if (v < lmin) { lmin = v; lidx = j; }
        }
      }
    }

    // softmax over the 32 lanes (K == wave32): w_k = softmax_k(-d2_k)
    float lg = -myv;
    float mx = lg;
#pragma unroll
    for (int off = 16; off >= 1; off >>= 1) mx = fmaxf(mx, __shfl_xor(mx, off, 32));
    float e = __expf(lg - mx);
    float esum = e;
#pragma unroll
    for (int off = 16; off >= 1; off >>= 1) esum += __shfl_xor(esum, off, 32);
    const float w = e / esum;

    // dbar[d] = sum_k w_k * (f[idx_k][d] - f_i[d])  (butterfly reduce -> all lanes)
    const float* fi = fb + (size_t)i * FPAD;
    const float* fj = fb + (size_t)myidx * FPAD;
    float dbar[FDIM];
#pragma unroll 1
    for (int d = 0; d < FDIM; ++d) {
      float cpart = w * (fj[d] - fi[d]);
#pragma unroll
      for (int off = 16; off >= 1; off >>= 1) cpart += __shfl_xor(cpart, off, 32);
      dbar[d] = cpart;
    }

    // gen[i][o] = relu(b1[o] + f_i . W1a[:,o] + dbar . W1b[:,o]); lane = output channel o
    const int o = lane;
    float acc = b1[o];
#pragma unroll 1
    for (int d = 0; d < FDIM; ++d) {
      acc += fi[d]   * W1[d * FIN + o];
      acc += dbar[d] * W1[(FDIM + d) * FIN + o];
    }
    float g = fmaxf(acc, 0.f);
    const size_t obase = ((size_t)b * NPT + i) * FIN;
    gen_out[obase + o] = g;

    float err = g - y[obase + o];
    err *= err;
#pragma unroll
    for (int off = 16; off >= 1; off >>= 1) err += __shfl_xor(err, off, 32);
    if (lane == 0) errs[m] = err;
  }
  __syncthreads();

  if (tid == 0) {
    float s = 0.f;
#pragma unroll
    for (int m = 0; m < ROWS; ++m) s += errs[m];
    partials[blockIdx.x] = s;
  }
}

// ---------------- Kernel 3: deterministic MSE reduction ----------------
__global__ void ama_finalize_kernel(const float* __restrict__ partials, int nparts,
                                    float* __restrict__ out_mse) {
  __shared__ float sbuf[256];
  int t = threadIdx.x;
  float a = 0.f;
  for (int p = t; p < nparts; p += 256) a += partials[p];
  sbuf[t] = a;
  __syncthreads();
  for (int off = 128; off > 0; off >>= 1) {
    if (t < off) sbuf[t] += sbuf[t + off];
    __syncthreads();
  }
  if (t == 0) out_mse[0] = sbuf[0] / (float)(B * NPT * FIN);
}

extern "C" void kernel_launch(void* const* d_in, const int* in_sizes, int n_in,
                              void* d_out, int out_size, void* d_ws, size_t ws_size,
                              hipStream_t stream) {
  const float* x     = (const float*)d_in[0];
  const float* noise = (const float*)d_in[1];
  const float* y     = (const float*)d_in[2];
  const float* W1    = (const float*)d_in[3];
  const float* b1    = (const float*)d_in[4];

  float* gen = (float*)d_out;                       // B*NPT*FIN floats
  float* mse = gen + (size_t)B * NPT * FIN;         // 1 float

  float* ws = (float*)d_ws;
  float* fpad = ws;                                  // B*NPT*FPAD = 1,310,720 floats
  float* sqn  = fpad + (size_t)B * NPT * FPAD;       // B*NPT = 32,768 floats
  float* parts = sqn + (size_t)B * NPT;              // 2048 floats

  const int npts = B * NPT;
  ama_prep_kernel<<<(npts + 255) / 256, 256, 0, stream>>>(x, noise, fpad, sqn);

  const int nblocks = B * (NPT / ROWS);              // 2048
  ama_knn_main_kernel<<<nblocks, 256, 0, stream>>>(fpad, sqn, y, W1, b1, gen, parts);

  ama_finalize_kernel<<<1, 256, 0, stream>>>(parts, nblocks, mse);
}